// RouterOursNewTokenReductionRatio_38036230373987
// MI455X (gfx1250) — compile-verified
//
#include <hip/hip_runtime.h>
#include <hip/hip_bf16.h>

// Problem constants (match the reference)
#define B_   8
#define L_   1024
#define D_   768
#define H_   12
#define HP_  16      // heads padded to 16 for WMMA A-matrix
#define HD_  64
#define NEGF (-3.4028234663852886e38f)   // float32 finfo.min

typedef float v2f __attribute__((ext_vector_type(2)));
typedef float v8f __attribute__((ext_vector_type(8)));

// ---------------------------------------------------------------------------
// 1) valid mask, valid count, K_list; also zero the imp accumulator
__global__ void k_mask_stats(const float* __restrict__ am,
                             float* __restrict__ valid, float* __restrict__ cnt,
                             int* __restrict__ Klist, float* __restrict__ imp) {
  int b = blockIdx.x, t = threadIdx.x;
  float a = am[b * L_ + t];
  float v = (a > -10.0f) ? 1.0f : 0.0f;
  valid[b * L_ + t] = v;
  imp[b * L_ + t] = 0.0f;
  __shared__ float red[L_];
  red[t] = v;
  __syncthreads();
  for (int s = L_ / 2; s > 0; s >>= 1) {
    if (t < s) red[t] += red[t + s];
    __syncthreads();
  }
  if (t == 0) {
    float c = red[0];
    cnt[b] = c;
    int K = (int)(c * 0.5f) - 1;   // (valid_len * R) trunc, minus num_new_token
    if (K < 1) K = 1;
    Klist[b] = K;
  }
}

// ---------------------------------------------------------------------------
// 2) sentence[b,:] = mean over valid tokens of hidden  (softmax of 0/-1e9 mask)
__global__ void k_sentence(const float* __restrict__ hid, const float* __restrict__ valid,
                           const float* __restrict__ cnt, float* __restrict__ sent) {
  int b = blockIdx.x, t = threadIdx.x;  // t in [0,768)
  __shared__ float vsh[L_];
  for (int i = t; i < L_; i += D_) vsh[i] = valid[b * L_ + i];
  __syncthreads();
  float acc = 0.0f;
  const float* hrow = hid + (size_t)b * L_ * D_ + t;
  for (int l = 0; l < L_; ++l) acc += vsh[l] * hrow[(size_t)l * D_];
  sent[b * D_ + t] = acc / cnt[b];
}

// ---------------------------------------------------------------------------
// 3) Q[b,:] = sentence[b,:] @ Wq + bq
__global__ void k_qproj(const float* __restrict__ sent, const float* __restrict__ Wq,
                        const float* __restrict__ bq, float* __restrict__ Q) {
  int b = blockIdx.x, t = threadIdx.x;
  __shared__ float s[D_];
  s[t] = sent[b * D_ + t];
  __syncthreads();
  float acc = bq[t];
  for (int d = 0; d < D_; ++d) acc += s[d] * Wq[(size_t)d * D_ + t];
  Q[b * D_ + t] = acc;
}

// ---------------------------------------------------------------------------
// 4) upad[b,h,d] = sum_j Wk[d, h*64+j] * Q[b, h*64+j]  for h<12, zeros for h>=12
//    cvec16[b,h] = Q_h . bk_h (zeros for h>=12)
//    Zero-padding here removes ALL divergence from the WMMA K-loop below.
__global__ void k_uproj(const float* __restrict__ Q, const float* __restrict__ Wk,
                        const float* __restrict__ bk, float* __restrict__ upad,
                        float* __restrict__ cvec16) {
  int h = blockIdx.x, b = blockIdx.y, t = threadIdx.x;  // h in [0,16)
  if (h >= H_) {
    upad[((size_t)b * HP_ + h) * D_ + t] = 0.0f;
    if (t == 0) cvec16[b * HP_ + h] = 0.0f;
    return;
  }
  __shared__ float qc[HD_], bkc[HD_];
  if (t < HD_) { qc[t] = Q[b * D_ + h * HD_ + t]; bkc[t] = bk[h * HD_ + t]; }
  __syncthreads();
  float acc = 0.0f;
  const float* wrow = Wk + (size_t)t * D_ + h * HD_;
  #pragma unroll 8
  for (int j = 0; j < HD_; ++j) acc += wrow[j] * qc[j];
  upad[((size_t)b * HP_ + h) * D_ + t] = acc;
  if (t == 0) {
    float cb = 0.0f;
    for (int j = 0; j < HD_; ++j) cb += qc[j] * bkc[j];
    cvec16[b * HP_ + h] = cb;
  }
}

// ---------------------------------------------------------------------------
// 5) imp[b,k] += sum over a 256-row chunk of (h,q) of sas[b,h,q,k]*m[b,q]
//    400 MB streaming reduction, fully coalesced (k contiguous per row).
__global__ void k_imp(const float* __restrict__ sas, const float* __restrict__ valid,
                      float* __restrict__ imp) {
  int b = blockIdx.y, chunk = blockIdx.x, t = threadIdx.x;  // 1024 threads
  __shared__ float vsh[L_];
  vsh[t] = valid[b * L_ + t];
  __syncthreads();
  float acc = 0.0f;
  int row0 = chunk * 256;
  for (int r = 0; r < 256; ++r) {
    int row = row0 + r;           // row in [0, 12*1024)
    int h = row >> 10;
    int q = row & (L_ - 1);
    if (vsh[q] != 0.0f) {
      acc += sas[((size_t)(b * H_ + h) * L_ + q) * L_ + t];
    }
  }
  atomicAdd(&imp[b * L_ + t], acc);
}

// ---------------------------------------------------------------------------
// 6) exact double-argsort top-K (stable ties by index), write mask + tome outputs
__global__ void k_topk(const float* __restrict__ imp, const float* __restrict__ valid,
                       const int* __restrict__ Klist, const float* __restrict__ am,
                       float* __restrict__ outMask, float* __restrict__ outTome) {
  int b = blockIdx.x, t = threadIdx.x;  // 1024 threads
  __shared__ float simp[L_];
  float x = imp[b * L_ + t] * valid[b * L_ + t] * (1.0f / (float)(H_ * L_));
  if (t == 0) x = __builtin_inff();     // always preserve CLS
  simp[t] = x;
  __syncthreads();
  int K = Klist[b];
  int rank = 0;
  for (int j = 0; j < L_; ++j) {
    float y = simp[j];
    rank += (y > x) || (y == x && j < t);
  }
  bool keep = rank < K;
  float amv = am[b * L_ + t];
  outMask[b * (L_ + 1) + t] = keep ? amv : NEGF;
  outTome[b * (L_ + 1) + t] = 1.0f;
  if (t == 0) {
    outMask[b * (L_ + 1) + L_] = 0.0f;   // appended new-token mask slot
    outTome[b * (L_ + 1) + L_] = 1.0f;
  }
}

// ---------------------------------------------------------------------------
// 7) WMMA f32 16x16x4: score[b,h,l] = (upad[b,h,:] . hidden[b,l,:] + cvec[b,h]) / 8
//    A = Upad (16x768, rows 12..15 are zero), B = hidden[b]^T (768x1024).
//    One wave per 16x16 output tile; branch-free K-loop of 192 WMMAs.
__global__ void k_score_wmma(const float* __restrict__ hid, const float* __restrict__ upad,
                             const float* __restrict__ cvec16, float* __restrict__ score) {
  int b = blockIdx.y;
  int wid = threadIdx.x >> 5;          // 4 waves per block
  int lane = threadIdx.x & 31;
  int tile = blockIdx.x * 4 + wid;     // 64 N-tiles per batch
  int col0 = tile * 16;
  int half = lane >> 4;                // 0: K=k0,k0+1   1: K=k0+2,k0+3
  int l16 = lane & 15;

  const float* aptr = upad + ((size_t)b * HP_ + l16) * D_ + 2 * half;            // A row m=l16
  const float* bptr = hid + ((size_t)b * L_ + (col0 + l16)) * D_ + 2 * half;     // B col n=l16

  v8f acc = {};
  #pragma unroll 4
  for (int k0 = 0; k0 < D_; k0 += 4) {
    v2f a, bb;
    a.x  = aptr[k0];
    a.y  = aptr[k0 + 1];
    bb.x = bptr[k0];
    bb.y = bptr[k0 + 1];
    acc = __builtin_amdgcn_wmma_f32_16x16x4_f32(
        /*neg_a=*/false, a, /*neg_b=*/false, bb,
        /*c_mod=*/(short)0, acc, /*reuse_a=*/false, /*reuse_b=*/false);
  }
  // C layout: lane<16 -> (M=r, N=lane); lane>=16 -> (M=r+8, N=lane-16)
  #pragma unroll
  for (int r = 0; r < 8; ++r) {
    int m = r + 8 * half;
    if (m < H_) {
      float cb = cvec16[b * HP_ + m];
      score[((size_t)b * H_ + m) * L_ + col0 + l16] = (acc[r] + cb) * 0.125f;
    }
  }
}

// ---------------------------------------------------------------------------
// 8) per (b,h) masked softmax over L, in place
__global__ void k_softmax(float* __restrict__ score, const float* __restrict__ valid) {
  int h = blockIdx.x, b = blockIdx.y, t = threadIdx.x;  // 1024 threads
  float* s = score + ((size_t)b * H_ + h) * L_;
  float v = valid[b * L_ + t];
  float x = (v != 0.0f) ? s[t] : NEGF;
  __shared__ float red[L_];
  red[t] = x;
  __syncthreads();
  for (int st = L_ / 2; st > 0; st >>= 1) {
    if (t < st) red[t] = fmaxf(red[t], red[t + st]);
    __syncthreads();
  }
  float mx = red[0];
  __syncthreads();
  float e = (v != 0.0f) ? expf(x - mx) : 0.0f;
  red[t] = e;
  __syncthreads();
  for (int st = L_ / 2; st > 0; st >>= 1) {
    if (t < st) red[t] += red[t + st];
    __syncthreads();
  }
  s[t] = e / red[0];
}

// ---------------------------------------------------------------------------
// 9) hbar[b,h,d] = sum_l w[b,h,l] * hidden[b,l,d]   (single pass over hidden)
__global__ void k_hbar(const float* __restrict__ hid, const float* __restrict__ w,
                       float* __restrict__ hbar) {
  int b = blockIdx.x, t = threadIdx.x;  // 768 threads
  __shared__ float wsh[H_ * L_];        // 48 KB
  for (int i = t; i < H_ * L_; i += D_) wsh[i] = w[(size_t)b * H_ * L_ + i];
  __syncthreads();
  float acc[H_];
  #pragma unroll
  for (int h = 0; h < H_; ++h) acc[h] = 0.0f;
  const float* hrow = hid + (size_t)b * L_ * D_ + t;
  for (int l = 0; l < L_; ++l) {
    float x = hrow[(size_t)l * D_];
    #pragma unroll
    for (int h = 0; h < H_; ++h) acc[h] += wsh[h * L_ + l] * x;
  }
  #pragma unroll
  for (int h = 0; h < H_; ++h) hbar[((size_t)b * H_ + h) * D_ + t] = acc[h];
}

// ---------------------------------------------------------------------------
// 10) attnout[b,t] = hbar[b, t/64, :] @ Wv[:, t] + bv[t]
__global__ void k_attnout(const float* __restrict__ hbar, const float* __restrict__ Wv,
                          const float* __restrict__ bv, float* __restrict__ attnout) {
  int b = blockIdx.x, t = threadIdx.x;
  int h = t >> 6;
  __shared__ float hb[H_ * D_];
  for (int i = t; i < H_ * D_; i += D_) hb[i] = hbar[(size_t)b * H_ * D_ + i];
  __syncthreads();
  float acc = bv[t];
  const float* hv = hb + h * D_;
  for (int d = 0; d < D_; ++d) acc += hv[d] * Wv[(size_t)d * D_ + t];
  attnout[b * D_ + t] = acc;
}

// ---------------------------------------------------------------------------
// 11) new_token row: out[b, 1024, :] = attnout[b,:] @ Wo + bo
__global__ void k_newtoken(const float* __restrict__ attnout, const float* __restrict__ Wo,
                           const float* __restrict__ bo, float* __restrict__ outTok) {
  int b = blockIdx.x, t = threadIdx.x;
  __shared__ float ao[D_];
  ao[t] = attnout[b * D_ + t];
  __syncthreads();
  float acc = bo[t];
  for (int uu = 0; uu < D_; ++uu) acc += ao[uu] * Wo[(size_t)uu * D_ + t];
  outTok[((size_t)b * (L_ + 1) + L_) * D_ + t] = acc;
}

// ---------------------------------------------------------------------------
// 12) copy hidden into rows 0..1023 of final_token (float4, coalesced)
__global__ void k_copy(const float* __restrict__ hid, float* __restrict__ outTok) {
  int i = blockIdx.x * blockDim.x + threadIdx.x;   // float4 index, B*L*192 total
  const int PER_ROW = D_ / 4;                      // 192
  int d4 = i % PER_ROW;
  int l = (i / PER_ROW) % L_;
  int b = i / (PER_ROW * L_);
  const float4* src = (const float4*)hid;
  float4* dst = (float4*)outTok;
  dst[((size_t)b * (L_ + 1) + l) * PER_ROW + d4] = src[(size_t)i];
}

// ---------------------------------------------------------------------------
extern "C" void kernel_launch(void* const* d_in, const int* in_sizes, int n_in,
                              void* d_out, int out_size, void* d_ws, size_t ws_size,
                              hipStream_t stream) {
  (void)in_sizes; (void)n_in; (void)out_size; (void)ws_size;
  const float* hid  = (const float*)d_in[0];
  const float* am   = (const float*)d_in[1];
  const float* sas  = (const float*)d_in[2];
  // d_in[3] key_layer, d_in[4] tome_size: unused by the reference forward
  const float* Wq   = (const float*)d_in[5];
  const float* bq   = (const float*)d_in[6];
  const float* Wk   = (const float*)d_in[7];
  const float* bk   = (const float*)d_in[8];
  const float* Wv   = (const float*)d_in[9];
  const float* bv   = (const float*)d_in[10];
  const float* Wo   = (const float*)d_in[11];
  const float* bo   = (const float*)d_in[12];

  float* out = (float*)d_out;
  float* outTok  = out;                                  // [B, L+1, D]
  float* outMask = out + (size_t)B_ * (L_ + 1) * D_;     // [B, 1, 1, L+1]
  float* outTome = outMask + (size_t)B_ * (L_ + 1);      // [B, L+1, 1]

  // workspace carve (floats)
  float* ws = (float*)d_ws;
  float* valid   = ws;                      // B*L         = 8192
  float* cnt     = valid + B_ * L_;         // B           = 8
  int*   Klist   = (int*)(cnt + B_);        // B           = 8
  float* sent    = (float*)(Klist + B_);    // B*D         = 6144
  float* Q       = sent + B_ * D_;          // B*D         = 6144
  float* upad    = Q + B_ * D_;             // B*16*D      = 98304 (heads zero-padded)
  float* cvec16  = upad + B_ * HP_ * D_;    // B*16        = 128
  float* imp     = cvec16 + B_ * HP_;       // B*L         = 8192
  float* score   = imp + B_ * L_;           // B*H*L       = 98304
  float* hbar    = score + B_ * H_ * L_;    // B*H*D       = 73728
  float* attnout = hbar + B_ * H_ * D_;     // B*D         = 6144  (~1.2 MB total)

  k_mask_stats<<<B_, L_, 0, stream>>>(am, valid, cnt, Klist, imp);
  k_sentence  <<<B_, D_, 0, stream>>>(hid, valid, cnt, sent);
  k_qproj     <<<B_, D_, 0, stream>>>(sent, Wq, bq, Q);
  k_uproj     <<<dim3(HP_, B_), D_, 0, stream>>>(Q, Wk, bk, upad, cvec16);
  k_imp       <<<dim3(48, B_), L_, 0, stream>>>(sas, valid, imp);
  k_topk      <<<B_, L_, 0, stream>>>(imp, valid, Klist, am, outMask, outTome);
  k_score_wmma<<<dim3(16, B_), 128, 0, stream>>>(hid, upad, cvec16, score);
  k_softmax   <<<dim3(H_, B_), L_, 0, stream>>>(score, valid);
  k_hbar      <<<B_, D_, 0, stream>>>(hid, score, hbar);
  k_attnout   <<<B_, D_, 0, stream>>>(hbar, Wv, bv, attnout);
  k_newtoken  <<<B_, D_, 0, stream>>>(attnout, Wo, bo, outTok);
  k_copy      <<<(B_ * L_ * (D_ / 4)) / 256, 256, 0, stream>>>(hid, outTok);
}